// GCNNet_30081950941674
// MI455X (gfx1250) — compile-verified
//
#include <hip/hip_runtime.h>
#include <math.h>

// ---------------------------------------------------------------------------
// GCN 2-layer forward for MI455X (gfx1250, wave32).
//   N=50000 nodes, E=800000 edges, C: 64 -> 64 -> 40.
// L2-resident problem (all tensors ~45MB << 192MB L2): scatter-bound.
// GEMMs: V_WMMA_F32_16X16X4_F32, A preloaded as b64 fragments (clamped,
// branchless), B from LDS-transposed W via ds_load_b64 (stride-66 padding).
// Scatter-add: relaxed agent-scope f32 atomics (global_atomic_add_f32).
// ---------------------------------------------------------------------------

typedef __attribute__((ext_vector_type(2))) float v2f;
typedef __attribute__((ext_vector_type(8))) float v8f;

#define CIN  64
#define CHID 64
#define COUT 40
#define LDSB 66   // LDS row stride (floats) for transposed W: avoids bank conflicts

static __device__ __forceinline__ void atomic_add_f32(float* p, float v) {
    __hip_atomic_fetch_add(p, v, __ATOMIC_RELAXED, __HIP_MEMORY_SCOPE_AGENT);
}

// ---- degree / normalization ----------------------------------------------
__global__ void k_init_deg(float* __restrict__ deg, int N) {
    int i = blockIdx.x * 256 + threadIdx.x;
    if (i < N) deg[i] = 1.0f;                       // self loop contributes 1
}

__global__ void k_accum_deg(const int* __restrict__ dst, float* __restrict__ deg, int E) {
    int e = blockIdx.x * 256 + threadIdx.x;
    if (e < E) atomic_add_f32(&deg[dst[e]], 1.0f);
}

__global__ void k_dinv(float* __restrict__ deg, int N) {
    int i = blockIdx.x * 256 + threadIdx.x;
    if (i < N) deg[i] = rsqrtf(deg[i]);             // deg >= 1 always
}

// ---- layer 1 GEMM: h1 = x @ W1 ; out1 = b1 + h1 * dinv^2 (self-loop term) --
// One wave per 16-row tile, 8 waves per block. wave32.
__global__ void k_gemm1_wmma(const float* __restrict__ x,
                             const float* __restrict__ W1,
                             const float* __restrict__ b1,
                             const float* __restrict__ dinv,
                             float* __restrict__ h1,
                             float* __restrict__ out1,
                             int N) {
    __shared__ float Wt[CHID * LDSB];               // Wt[c*LDSB + k] = W1[k][c]
    for (int idx = threadIdx.x; idx < CIN * CHID; idx += 256) {
        const int k = idx >> 6, c = idx & 63;
        Wt[c * LDSB + k] = W1[idx];
    }
    __syncthreads();

    const int lane  = threadIdx.x & 31;
    const int wave  = threadIdx.x >> 5;
    const int tile  = blockIdx.x * 8 + wave;
    const int tiles = (N + 15) >> 4;
    if (tile < tiles) {                             // wave-uniform guard (after barrier)
        const int rl   = lane & 15;                 // A row / B,C column within 16-tile
        const int half = lane >> 4;                 // K-pair selector
        const int row  = tile * 16 + rl;
        const int ar   = (row < N) ? row : 0;       // clamped, branchless; junk rows never stored

        // Preload all 16 A fragments: lane takes K pairs {kb+2h, kb+2h+1}, kb=0,4,...,60
        v2f afrag[16];
        const float* xr = x + (size_t)ar * CIN + 2 * half;
#pragma unroll
        for (int j = 0; j < 16; ++j)
            afrag[j] = *(const v2f*)(xr + 4 * j);   // global_load_b64

        v8f acc0 = {}, acc1 = {}, acc2 = {}, acc3 = {};
#pragma unroll
        for (int j = 0; j < 16; ++j) {
            const int k0 = 4 * j + 2 * half;
            const v2f b0 = *(const v2f*)(&Wt[( 0 + rl) * LDSB + k0]);  // ds_load_b64
            const v2f b1v= *(const v2f*)(&Wt[(16 + rl) * LDSB + k0]);
            const v2f b2 = *(const v2f*)(&Wt[(32 + rl) * LDSB + k0]);
            const v2f b3 = *(const v2f*)(&Wt[(48 + rl) * LDSB + k0]);
            acc0 = __builtin_amdgcn_wmma_f32_16x16x4_f32(false, afrag[j], false, b0,  (short)0, acc0, false, false);
            acc1 = __builtin_amdgcn_wmma_f32_16x16x4_f32(false, afrag[j], false, b1v, (short)0, acc1, false, false);
            acc2 = __builtin_amdgcn_wmma_f32_16x16x4_f32(false, afrag[j], false, b2,  (short)0, acc2, false, false);
            acc3 = __builtin_amdgcn_wmma_f32_16x16x4_f32(false, afrag[j], false, b3,  (short)0, acc3, false, false);
        }

#pragma unroll
        for (int r = 0; r < 8; ++r) {
            const int m = tile * 16 + r + 8 * half; // C/D layout: VGPR r -> M = r + 8*(lane/16)
            if (m < N) {
                const float dv  = dinv[m];
                const float dv2 = dv * dv;
                { const int c =  0 + rl; const float v = acc0[r];
                  h1[(size_t)m * CHID + c] = v; out1[(size_t)m * CHID + c] = b1[c] + v * dv2; }
                { const int c = 16 + rl; const float v = acc1[r];
                  h1[(size_t)m * CHID + c] = v; out1[(size_t)m * CHID + c] = b1[c] + v * dv2; }
                { const int c = 32 + rl; const float v = acc2[r];
                  h1[(size_t)m * CHID + c] = v; out1[(size_t)m * CHID + c] = b1[c] + v * dv2; }
                { const int c = 48 + rl; const float v = acc3[r];
                  h1[(size_t)m * CHID + c] = v; out1[(size_t)m * CHID + c] = b1[c] + v * dv2; }
            }
        }
    }
}

// ---- layer 1 scatter: out1[dst] += h1[src] * dinv[src]*dinv[dst], 64 ch ----
__global__ void k_scatter64(const int* __restrict__ src, const int* __restrict__ dst,
                            const float* __restrict__ dinv,
                            const float* __restrict__ h1,
                            float* __restrict__ out1, int E) {
    const long long tid = (long long)blockIdx.x * 256 + threadIdx.x;
    const long long e   = tid >> 5;
    const int lane      = (int)(tid & 31);
    if (e < E) {
        const int s = src[e], d = dst[e];
        const float nrm = dinv[s] * dinv[d];
        atomic_add_f32(&out1[(size_t)d * CHID + lane],      h1[(size_t)s * CHID + lane]      * nrm);
        atomic_add_f32(&out1[(size_t)d * CHID + lane + 32], h1[(size_t)s * CHID + lane + 32] * nrm);
    }
}

// ---- layer 2 GEMM: h2 = relu(out1) @ W2 ; out = b2 + h2 * dinv^2 -----------
__global__ void k_gemm2_wmma(const float* __restrict__ out1,
                             const float* __restrict__ W2,
                             const float* __restrict__ b2,
                             const float* __restrict__ dinv,
                             float* __restrict__ h2,
                             float* __restrict__ out,
                             int N) {
    __shared__ float Wt[48 * LDSB];                 // Wt[c*LDSB + k] = (c<40) ? W2[k][c] : 0
    for (int idx = threadIdx.x; idx < 48 * CHID; idx += 256) {
        const int c = idx >> 6, k = idx & 63;
        Wt[c * LDSB + k] = (c < COUT) ? W2[(size_t)k * COUT + c] : 0.0f;
    }
    __syncthreads();

    const int lane  = threadIdx.x & 31;
    const int wave  = threadIdx.x >> 5;
    const int tile  = blockIdx.x * 8 + wave;
    const int tiles = (N + 15) >> 4;
    if (tile < tiles) {
        const int rl   = lane & 15;
        const int half = lane >> 4;
        const int row  = tile * 16 + rl;
        const int ar   = (row < N) ? row : 0;

        // Preload + fuse ReLU into A fragments
        v2f afrag[16];
        const float* xr = out1 + (size_t)ar * CHID + 2 * half;
#pragma unroll
        for (int j = 0; j < 16; ++j) {
            v2f a = *(const v2f*)(xr + 4 * j);
            a.x = fmaxf(a.x, 0.0f); a.y = fmaxf(a.y, 0.0f);
            afrag[j] = a;
        }

        v8f acc0 = {}, acc1 = {}, acc2 = {};        // 48 cols (COUT=40 zero-padded)
#pragma unroll
        for (int j = 0; j < 16; ++j) {
            const int k0 = 4 * j + 2 * half;
            const v2f b0 = *(const v2f*)(&Wt[( 0 + rl) * LDSB + k0]);
            const v2f b1v= *(const v2f*)(&Wt[(16 + rl) * LDSB + k0]);
            const v2f b2v= *(const v2f*)(&Wt[(32 + rl) * LDSB + k0]);
            acc0 = __builtin_amdgcn_wmma_f32_16x16x4_f32(false, afrag[j], false, b0,  (short)0, acc0, false, false);
            acc1 = __builtin_amdgcn_wmma_f32_16x16x4_f32(false, afrag[j], false, b1v, (short)0, acc1, false, false);
            acc2 = __builtin_amdgcn_wmma_f32_16x16x4_f32(false, afrag[j], false, b2v, (short)0, acc2, false, false);
        }

#pragma unroll
        for (int r = 0; r < 8; ++r) {
            const int m = tile * 16 + r + 8 * half;
            if (m < N) {
                const float dv  = dinv[m];
                const float dv2 = dv * dv;
                { const int c =  0 + rl; const float v = acc0[r];
                  h2[(size_t)m * COUT + c] = v; out[(size_t)m * COUT + c] = b2[c] + v * dv2; }
                { const int c = 16 + rl; const float v = acc1[r];
                  h2[(size_t)m * COUT + c] = v; out[(size_t)m * COUT + c] = b2[c] + v * dv2; }
                { const int c = 32 + rl;
                  if (c < COUT) { const float v = acc2[r];
                    h2[(size_t)m * COUT + c] = v; out[(size_t)m * COUT + c] = b2[c] + v * dv2; } }
            }
        }
    }
}

// ---- layer 2 scatter: out[dst] += h2[src] * norm, 40 channels --------------
__global__ void k_scatter40(const int* __restrict__ src, const int* __restrict__ dst,
                            const float* __restrict__ dinv,
                            const float* __restrict__ h2,
                            float* __restrict__ out, int E) {
    const long long tid = (long long)blockIdx.x * 256 + threadIdx.x;
    const long long e   = tid >> 5;
    const int lane      = (int)(tid & 31);
    if (e < E) {
        const int s = src[e], d = dst[e];
        const float nrm = dinv[s] * dinv[d];
        atomic_add_f32(&out[(size_t)d * COUT + lane], h2[(size_t)s * COUT + lane] * nrm);
        if (lane < COUT - 32)
            atomic_add_f32(&out[(size_t)d * COUT + lane + 32], h2[(size_t)s * COUT + lane + 32] * nrm);
    }
}

// ---- in-place row-wise log_softmax over 40 channels ------------------------
__global__ void k_logsoftmax(float* __restrict__ out, int N) {
    const int i = blockIdx.x * 256 + threadIdx.x;
    if (i >= N) return;
    float* p = out + (size_t)i * COUT;
    float mx = p[0];
#pragma unroll
    for (int c = 1; c < COUT; ++c) mx = fmaxf(mx, p[c]);
    float s = 0.0f;
#pragma unroll
    for (int c = 0; c < COUT; ++c) s += expf(p[c] - mx);
    const float ls = logf(s);
#pragma unroll
    for (int c = 0; c < COUT; ++c) p[c] = p[c] - mx - ls;
}

// ---------------------------------------------------------------------------
extern "C" void kernel_launch(void* const* d_in, const int* in_sizes, int n_in,
                              void* d_out, int out_size, void* d_ws, size_t ws_size,
                              hipStream_t stream) {
    const float* x  = (const float*)d_in[0];
    const int*   ei = (const int*)d_in[1];        // [2, E] flat: src row then dst row
    const float* W1 = (const float*)d_in[2];
    const float* b1 = (const float*)d_in[3];
    const float* W2 = (const float*)d_in[4];
    const float* b2 = (const float*)d_in[5];

    const int E = in_sizes[1] / 2;
    const int N = in_sizes[0] / CIN;
    const int* src = ei;
    const int* dst = ei + E;

    // workspace layout (floats): dinv | h1 | out1 | h2   (~33.8 MB total)
    float* dinv = (float*)d_ws;
    float* h1   = dinv + (((size_t)N + 63) & ~(size_t)63);
    float* out1 = h1 + (size_t)N * CHID;
    float* h2   = out1 + (size_t)N * CHID;
    float* out  = (float*)d_out;

    const int tN = (N + 255) / 256;
    const int tE = (E + 255) / 256;

    k_init_deg <<<tN, 256, 0, stream>>>(dinv, N);
    k_accum_deg<<<tE, 256, 0, stream>>>(dst, dinv, E);
    k_dinv     <<<tN, 256, 0, stream>>>(dinv, N);

    const int tiles  = (N + 15) / 16;
    const int gblk   = (tiles + 7) / 8;            // 8 waves (16-row tiles) per block
    k_gemm1_wmma<<<gblk, 256, 0, stream>>>(x, W1, b1, dinv, h1, out1, N);

    const long long sth = (long long)E * 32;       // one wave per edge
    const int sblk = (int)((sth + 255) / 256);
    k_scatter64<<<sblk, 256, 0, stream>>>(src, dst, dinv, h1, out1, E);

    k_gemm2_wmma<<<gblk, 256, 0, stream>>>(out1, W2, b2, dinv, h2, out, N);
    k_scatter40<<<sblk, 256, 0, stream>>>(src, dst, dinv, h2, out, E);

    k_logsoftmax<<<tN, 256, 0, stream>>>(out, N);
}